// RPN_27539330302080
// MI455X (gfx1250) — compile-verified
//
#include <hip/hip_runtime.h>
#include <math.h>

typedef unsigned int u32;
typedef unsigned long long u64;
typedef __attribute__((ext_vector_type(2))) float v2f;
typedef __attribute__((ext_vector_type(8))) float v8f;

#define KSEL 1000
#define KPAD 1024
#define NTILE 63              // ceil(1008/16)
#define ROWS (NTILE*16)      // 1008 padded rows
#define NMS_TH 0.7f
#define ONE_PLUS_TH 1.7f
#define TH_EPS 7.0e-10f       // 0.7 * 1e-9
#define IMG_WF 800.0f
#define IMG_HF 800.0f
#define SCALE_CLAMP_F 4.1351665567423560f   // log(1000/16)

#if defined(__AMDGCN__) && __has_builtin(__builtin_amdgcn_wmma_f32_16x16x4_f32)
#define USE_WMMA 1
#else
#define USE_WMMA 0
#endif

// monotonic float -> uint key (ascending)
__device__ __forceinline__ u32 f2key(float f){
  u32 u = __float_as_uint(f);
  return u ^ ((u >> 31) ? 0xFFFFFFFFu : 0x80000000u);
}

// ---------------- stage 0: deterministic scratch init ----------------
__global__ void k_clear(u32* __restrict__ hist, u32* __restrict__ prefix, u32* __restrict__ need,
                        u32* __restrict__ cnt, u32* __restrict__ ctie, u64* __restrict__ sel, int N){
  int t = blockIdx.x * blockDim.x + threadIdx.x;
  if (t < N*256) hist[t] = 0u;
  if (t < N){ prefix[t]=0u; need[t]=KSEL; cnt[t]=0u; ctie[t]=0u; }
  if (t < N*KPAD) sel[t] = 0ull;
}

// ---------------- stage 1: MSB-first radix select over 600k scores ----------------
__global__ void k_hist(const float* __restrict__ logits, int A, int r,
                       const u32* __restrict__ prefix, u32* __restrict__ hist){
  __shared__ u32 lh[256];
  int n = blockIdx.y;
  lh[threadIdx.x] = 0u;
  __syncthreads();
  int i = blockIdx.x*blockDim.x + threadIdx.x;
  if (i < A){
    const float* src = logits + (size_t)n*A;
    if (i + 16384 < A) __builtin_prefetch(src + i + 16384, 0, 0);  // global_prefetch_b8
    u32 key = f2key(src[i]);
    int shift = 8*(3-r);
    bool match = true;
    if (r > 0) match = ((key >> (shift+8)) == (prefix[n] >> (shift+8)));
    if (match) atomicAdd(&lh[(key>>shift)&255u], 1u);
  }
  __syncthreads();
  if (lh[threadIdx.x]) atomicAdd(&hist[n*256+threadIdx.x], lh[threadIdx.x]);
}

__global__ void k_scan(u32* __restrict__ hist, u32* __restrict__ prefix, u32* __restrict__ need, int r){
  __shared__ u32 h[256];
  int n = blockIdx.x;
  h[threadIdx.x] = hist[n*256+threadIdx.x];
  __syncthreads();
  hist[n*256+threadIdx.x] = 0u;   // re-zero for next round
  if (threadIdx.x == 0){
    u32 nd = need[n], acc = 0u; int b;
    for (b = 255; b > 0; --b){
      if (acc + h[b] >= nd) break;
      acc += h[b];
    }
    need[n]  = nd - acc;
    prefix[n] |= ((u32)b) << (8*(3-r));
  }
}

__global__ void k_collect(const float* __restrict__ logits, int A,
                          const u32* __restrict__ prefix, const u32* __restrict__ need,
                          u32* __restrict__ cnt, u32* __restrict__ ctie, u64* __restrict__ sel){
  int n = blockIdx.y;
  int i = blockIdx.x*blockDim.x + threadIdx.x;
  if (i >= A) return;
  u32 key = f2key(logits[(size_t)n*A + i]);
  u32 T = prefix[n];
  int slot = -1;
  if (key > T){
    slot = (int)atomicAdd(&cnt[n], 1u);
  } else if (key == T){
    u32 q = atomicAdd(&ctie[n], 1u);
    if (q < need[n]) slot = (int)atomicAdd(&cnt[n], 1u);
  }
  if (slot >= 0 && slot < KPAD)
    sel[(size_t)n*KPAD + slot] = ((u64)key << 32) | (u64)(0xFFFFFFFFu - (u32)i);
}

// ---------------- stage 2: bitonic sort (desc, tie = low index) + decode/clip ----------------
__global__ void __launch_bounds__(1024) k_sortdecode(
    const float* __restrict__ anchors, const float* __restrict__ logits,
    const float* __restrict__ deltas, int A, const u64* __restrict__ sel,
    float* __restrict__ boxes, float* __restrict__ scores, u32* __restrict__ validf){
  __shared__ u64 s[KPAD];
  int n = blockIdx.x, t = threadIdx.x;
  s[t] = sel[(size_t)n*KPAD + t];
  __syncthreads();
  for (int k = 2; k <= KPAD; k <<= 1){
    for (int j = k >> 1; j > 0; j >>= 1){
      int ixj = t ^ j;
      if (ixj > t){
        u64 a = s[t], b = s[ixj];
        bool desc = ((t & k) == 0);
        if (desc ? (a < b) : (a > b)){ s[t] = b; s[ixj] = a; }
      }
      __syncthreads();
    }
  }
  if (t < KSEL){
    u64 e = s[t];
    int idx = (int)(0xFFFFFFFFu - (u32)(e & 0xFFFFFFFFull));
    if (idx < 0 || idx >= A) idx = 0;               // pad safety
    float sc = logits[(size_t)n*A + idx];
    float ax1 = anchors[(size_t)idx*4+0], ay1 = anchors[(size_t)idx*4+1];
    float ax2 = anchors[(size_t)idx*4+2], ay2 = anchors[(size_t)idx*4+3];
    float w = ax2-ax1, h = ay2-ay1;
    float cx = ax1 + 0.5f*w, cy = ay1 + 0.5f*h;
    const float* d = deltas + ((size_t)n*A + idx)*4;
    float dx=d[0], dy=d[1];
    float dw=fminf(d[2], SCALE_CLAMP_F), dh=fminf(d[3], SCALE_CLAMP_F);
    float pcx = dx*w + cx, pcy = dy*h + cy;
    float pw = expf(dw)*w, ph = expf(dh)*h;
    float x1 = pcx - 0.5f*pw, y1 = pcy - 0.5f*ph;
    float x2 = pcx + 0.5f*pw, y2 = pcy + 0.5f*ph;
    x1 = fminf(fmaxf(x1, 0.f), IMG_WF); x2 = fminf(fmaxf(x2, 0.f), IMG_WF);
    y1 = fminf(fmaxf(y1, 0.f), IMG_HF); y2 = fminf(fmaxf(y2, 0.f), IMG_HF);
    bool fin = __builtin_isfinite(sc);
    float* b = boxes + ((size_t)n*KSEL + t)*4;
    b[0]=x1; b[1]=y1; b[2]=x2; b[3]=y2;
    scores[(size_t)n*KSEL + t] = sc;
    validf[(size_t)n*KSEL + t] = ((x2 > x1) && (y2 > y1) && fin) ? 1u : 0u;
  }
}

// ---------------- stage 3: WMMA-tiled NMS + stable partition + output ----------------
__global__ void __launch_bounds__(1024) k_nms(
    const float* __restrict__ boxes, const float* __restrict__ scores,
    const u32* __restrict__ validf, u32* __restrict__ rowmask, float* __restrict__ out){
  __shared__ float bx1[ROWS], by1[ROWS], bx2[ROWS], by2[ROWS];
  __shared__ float s_area[ROWS];
  __shared__ u32 s_scan[KPAD];
  __shared__ u32 s_keepw[32];

  int n = blockIdx.x, t = threadIdx.x;
  u32* rm = rowmask + (size_t)n * KPAD * 32;

  for (int i = t; i < ROWS; i += 1024){
    float x1=0.f,y1=0.f,x2=0.f,y2=0.f;
    if (i < KSEL){
      const float* b = boxes + ((size_t)n*KSEL + i)*4;
      x1=b[0]; y1=b[1]; x2=b[2]; y2=b[3];
    }
    bx1[i]=x1; by1[i]=y1; bx2[i]=x2; by2[i]=y2;
    s_area[i] = (x2-x1)*(y2-y1);
  }
  s_scan[t] = (t < KSEL) ? validf[(size_t)n*KSEL + t] : 0u;   // valid flags for the sweep
  for (int i = t; i < KPAD*32; i += 1024) rm[i] = 0u;
  __syncthreads();

  // --- pairwise suppression bitmasks, one 16x16 tile per wave per iteration ---
  // WMMA computes thsum[M][N] = NMS_TH * (area_i[M] + area_j[N]) on the matrix pipe:
  //   A(16x4): col0 = 0.7*area_i, col1 = 0.7 ; B(4x16): row0 = 1, row1 = area_j
  // Predicate (division-free, union > 0 always):
  //   inter/union > th  <=>  (1+th)*inter > th*(asum + eps)  <=>  1.7*inter > thsum + 0.7e-9
  int wid = t >> 5, lane = t & 31, half = lane >> 4, l = lane & 15;
  for (int tile = wid; tile < NTILE*NTILE; tile += 32){
    int ti = tile / NTILE, tj = tile % NTILE;
    int i0 = ti*16, j0 = tj*16;
    v8f thsum = {0.f,0.f,0.f,0.f,0.f,0.f,0.f,0.f};
#if USE_WMMA
    {
      v2f av, bv;
      av[0] = half ? 0.f : NMS_TH * s_area[i0 + l];
      av[1] = half ? 0.f : NMS_TH;
      bv[0] = half ? 0.f : 1.f;
      bv[1] = half ? 0.f : s_area[j0 + l];
      v8f cz = {0.f,0.f,0.f,0.f,0.f,0.f,0.f,0.f};
      thsum = __builtin_amdgcn_wmma_f32_16x16x4_f32(false, av, false, bv,
                                                    (short)0, cz, false, false);
    }
#else
    for (int v = 0; v < 8; ++v)
      thsum[v] = NMS_TH * (s_area[i0 + v + 8*half] + s_area[j0 + l]);
#endif
    int jj = j0 + l;                       // C/D layout: column N = lane & 15
    float jx1=bx1[jj], jy1=by1[jj], jx2=bx2[jj], jy2=by2[jj];
    for (int v = 0; v < 8; ++v){
      int ii = i0 + v + 8*half;            // C/D layout: row M = v + 8*(lane>=16)
      float ltx = fmaxf(bx1[ii], jx1), lty = fmaxf(by1[ii], jy1);
      float rbx = fminf(bx2[ii], jx2), rby = fminf(by2[ii], jy2);
      float iw = fmaxf(rbx-ltx, 0.f), ih = fmaxf(rby-lty, 0.f);
      float inter = iw*ih;
      bool pred = (ONE_PLUS_TH*inter > thsum[v] + TH_EPS)
               && (jj > ii) && (jj < KSEL) && (ii < KSEL);
      u32 m = (u32)__ballot(pred);         // wave32: bits[15:0] row ii(v), bits[31:16] row v+8
      if (lane == 0 && m){
        u32 w  = (u32)(tj >> 1);
        u32 sh = (u32)(tj & 1) * 16u;
        if (m & 0xFFFFu) atomicOr(&rm[(size_t)(i0+v  )*32 + w], (m & 0xFFFFu) << sh);
        if (m >> 16)     atomicOr(&rm[(size_t)(i0+v+8)*32 + w], (m >> 16)     << sh);
      }
    }
  }
  __syncthreads();

  // --- greedy sequential sweep, one wave; lane w owns removed-bits word w ---
  if (t < 32){
    u32 rem = 0u, kw = 0u;
    for (int i = 0; i < KSEL; ++i){
      u32 row = __hip_atomic_load(&rm[(size_t)i*32 + t], __ATOMIC_RELAXED,
                                  __HIP_MEMORY_SCOPE_AGENT);
      u32 rw = (u32)__shfl((int)rem, i >> 5);
      bool kp = (s_scan[i] != 0u) && !((rw >> (i & 31)) & 1u);
      if (kp){
        rem |= row;
        if ((i >> 5) == t) kw |= 1u << (i & 31);
      }
    }
    s_keepw[t] = kw;
  }
  __syncthreads();

  // --- stable partition: kept (in score order) then suppressed with -inf scores ---
  bool keep = (t < KSEL) && ((s_keepw[t >> 5] >> (t & 31)) & 1u);
  s_scan[t] = keep ? 1u : 0u;
  __syncthreads();
  for (int off = 1; off < KPAD; off <<= 1){
    u32 x = s_scan[t] + ((t >= off) ? s_scan[t - off] : 0u);
    __syncthreads();
    s_scan[t] = x;
    __syncthreads();
  }
  u32 total = s_scan[KSEL-1];
  if (t < KSEL){
    u32 incl = s_scan[t];
    u32 pos = keep ? (incl - 1u) : (total + (u32)(t + 1) - incl - 1u);
    float sc = keep ? scores[(size_t)n*KSEL + t] : -__builtin_inff();
    float* o = out + ((size_t)n*KSEL + pos)*5;
    o[0]=bx1[t]; o[1]=by1[t]; o[2]=bx2[t]; o[3]=by2[t]; o[4]=sc;
  }
}

// ---------------- launcher ----------------
extern "C" void kernel_launch(void* const* d_in, const int* in_sizes, int n_in,
                              void* d_out, int out_size, void* d_ws, size_t ws_size,
                              hipStream_t stream) {
  const float* anchors = (const float*)d_in[0];
  const float* logits  = (const float*)d_in[1];
  const float* deltas  = (const float*)d_in[2];
  float* out = (float*)d_out;

  int A = in_sizes[0] / 4;       // 600000 anchors
  int N = in_sizes[1] / A;       // 8 images

  char* p = (char*)d_ws;
  auto alloc = [&](size_t bytes)->void*{
    void* r = (void*)p; p += (bytes + 255) & ~(size_t)255; return r;
  };
  u32* hist     = (u32*)alloc((size_t)N*256*4);
  u32* prefix   = (u32*)alloc((size_t)N*4);
  u32* need     = (u32*)alloc((size_t)N*4);
  u32* cnt      = (u32*)alloc((size_t)N*4);
  u32* ctie     = (u32*)alloc((size_t)N*4);
  u64* sel      = (u64*)alloc((size_t)N*KPAD*8);
  float* boxes  = (float*)alloc((size_t)N*KSEL*4*4);
  float* scores = (float*)alloc((size_t)N*KSEL*4);
  u32* validf   = (u32*)alloc((size_t)N*KSEL*4);
  u32* rowmask  = (u32*)alloc((size_t)N*KPAD*32*4);

  int blksA = (A + 255) / 256;

  k_clear<<<dim3((N*KPAD + 255)/256), 256, 0, stream>>>(hist, prefix, need, cnt, ctie, sel, N);
  for (int r = 0; r < 4; ++r){
    k_hist<<<dim3(blksA, N), 256, 0, stream>>>(logits, A, r, prefix, hist);
    k_scan<<<dim3(N), 256, 0, stream>>>(hist, prefix, need, r);
  }
  k_collect<<<dim3(blksA, N), 256, 0, stream>>>(logits, A, prefix, need, cnt, ctie, sel);
  k_sortdecode<<<dim3(N), 1024, 0, stream>>>(anchors, logits, deltas, A, sel, boxes, scores, validf);
  k_nms<<<dim3(N), 1024, 0, stream>>>(boxes, scores, validf, rowmask, out);
}